// MultiBiasEncoder_23227183137572
// MI455X (gfx1250) — compile-verified
//
#include <hip/hip_runtime.h>

// MultiBiasEncoder, algebraically collapsed:
//   enc[l,s,c] = d*ws[s,c] + bs[s,c]*scale[s]  is affine in d = x - loc[l],
//   so LayerNorm stats are closed-form:  var(d,s) = A[s]d^2 + 2B[s]d + C[s].
//   out[tok,:] = [alpha(9) | beta(9) | loc_w(5) | 0] (1x24)  x  Basis (24x64)
// The 16384x24x64 GEMM runs on v_wmma_f32_16x16x4_f32 (wave32, K padded to 24).

typedef __attribute__((ext_vector_type(2))) float v2f;
typedef __attribute__((ext_vector_type(8))) float v8f;

#define EPSV    1e-6f
#define LN_EPSV 1e-5f

constexpr int Lc = 5;          // locations
constexpr int Sc = 9;          // scales
constexpr int Cc = 64;         // channels
constexpr int Kp = 24;         // padded basis rows: 9 alpha + 9 beta + 5 loc + 1 zero
constexpr int KSTR = 26;       // LDS row stride (keeps 8B alignment, spreads banks)
constexpr int TOK_PER_WG = 128;

__global__ __launch_bounds__(TOK_PER_WG)
void mbe_wmma_kernel(const float* __restrict__ x,
                     const float* __restrict__ ws,
                     const float* __restrict__ bs,
                     const float* __restrict__ loc_emb,
                     const float* __restrict__ scales,
                     const float* __restrict__ locs,
                     float* __restrict__ out, int BT)
{
    __shared__ float sB[Cc][KSTR];            // Basis^T: [channel][k], k in [0,24)
    __shared__ float sCf[TOK_PER_WG][KSTR];   // per-token coefficients [tok][k]
    __shared__ float sMw[Sc], sMb[Sc];        // row means of ws, bs*scale
    __shared__ float sA[Sc], sBq[Sc], sCq[Sc];// LayerNorm variance quadratic coeffs
    __shared__ float sScale[Sc], sRScale[Sc], sLoc[Lc];

    const int tid = threadIdx.x;

    // ---- Phase A0: per-scale means (9 threads; tiny) ----
    if (tid < Sc) {
        float sc = scales[tid];
        sScale[tid]  = sc;
        sRScale[tid] = 1.0f / sc;
        float mw = 0.f, mb = 0.f;
        for (int c = 0; c < Cc; ++c) { mw += ws[tid * Cc + c]; mb += bs[tid * Cc + c]; }
        sMw[tid] = mw * (1.0f / Cc);
        sMb[tid] = mb * sc * (1.0f / Cc);
    }
    if (tid < Lc) sLoc[tid] = locs[tid];
    __syncthreads();

    // ---- Phase A1: build centered basis^T in LDS (all 128 threads) ----
    for (int idx = tid; idx < Kp * Cc; idx += TOK_PER_WG) {
        int k = idx / Cc, n = idx % Cc;
        float v = 0.f;
        if (k < Sc)                 v = ws[k * Cc + n] - sMw[k];
        else if (k < 2 * Sc)        { int s = k - Sc; v = bs[s * Cc + n] * sScale[s] - sMb[s]; }
        else if (k < 2 * Sc + Lc)   v = loc_emb[(k - 2 * Sc) * Cc + n];
        sB[n][k] = v;
    }
    __syncthreads();

    // ---- Phase A2: variance quadratic coefficients A,B,C per scale ----
    if (tid < Sc) {
        float a = 0.f, b = 0.f, c2 = 0.f;
        for (int n = 0; n < Cc; ++n) {
            float wt = sB[n][tid];
            float bt = sB[n][tid + Sc];
            a += wt * wt; b += wt * bt; c2 += bt * bt;
        }
        sA[tid] = a * (1.0f / Cc);
        sBq[tid] = b * (1.0f / Cc);
        sCq[tid] = c2 * (1.0f / Cc);
    }
    __syncthreads();

    // ---- Phase B: per-token coefficient row (one thread = one token) ----
    const int tok = blockIdx.x * TOK_PER_WG + tid;
    {
        float xv = (tok < BT) ? x[tok] : 0.f;
        float d[Lc], lw[Lc];
        float lwsum = 0.f;
        #pragma unroll
        for (int l = 0; l < Lc; ++l) {
            d[l] = xv - sLoc[l];
            float w = 1.0f / (__logf(fabsf(d[l]) + 1.0f) + EPSV);
            lw[l] = w; lwsum += w;
        }
        float inv_lwsum = 1.0f / lwsum;

        float alpha[Sc], beta[Sc];
        #pragma unroll
        for (int s = 0; s < Sc; ++s) { alpha[s] = 0.f; beta[s] = 0.f; }

        #pragma unroll
        for (int l = 0; l < Lc; ++l) {
            float ad = fabsf(d[l]);
            float row[Sc]; float rs = 0.f;
            #pragma unroll
            for (int s = 0; s < Sc; ++s) {
                float t = 1.0f / (fabsf(__logf(ad * sRScale[s] + EPSV)) + EPSV);
                row[s] = t; rs += t;
            }
            float lwn = lw[l] * inv_lwsum / rs;   // loc weight * scale-row normalizer
            #pragma unroll
            for (int s = 0; s < Sc; ++s) {
                float var  = (sA[s] * d[l] + 2.0f * sBq[s]) * d[l] + sCq[s] + LN_EPSV;
                float invn = __frsqrt_rn(var);
                float w    = lwn * row[s] * invn;
                alpha[s] += w * d[l];
                beta[s]  += w;
            }
        }

        #pragma unroll
        for (int s = 0; s < Sc; ++s) { sCf[tid][s] = alpha[s]; sCf[tid][Sc + s] = beta[s]; }
        #pragma unroll
        for (int l = 0; l < Lc; ++l)  sCf[tid][2 * Sc + l] = lw[l] * inv_lwsum;
        sCf[tid][23] = 0.f;            // K padding row
    }
    __syncthreads();

    // ---- Phase C: (16x24)*(24x64) per token-tile via v_wmma_f32_16x16x4_f32 ----
    // wave32: 4 waves, each owns 2 token-tiles of 16 tokens x all 4 channel-tiles.
    const int wave  = tid >> 5;
    const int lane  = tid & 31;
    const int lhalf = lane >> 4;     // 0 or 1
    const int lmod  = lane & 15;

    // Uniform (scalar) predicate: is this whole block in range?  BT is a
    // multiple of TOK_PER_WG for the reference shapes, so this is the hot path
    // and avoids per-element saveexec/branch store tails entirely.
    const bool fullblock = (int)((blockIdx.x + 1) * TOK_PER_WG) <= BT;

    for (int tt = wave * 2; tt < wave * 2 + 2; ++tt) {
        // A fragments: 16x4 f32 tile per kb; VGPR v, lanes[16:31] hold K = k0+2+v
        v2f afrag[6];
        #pragma unroll
        for (int kb = 0; kb < 6; ++kb) {
            int k0 = 4 * kb + 2 * lhalf;
            const float* p = &sCf[tt * 16 + lmod][k0];   // 8B-aligned (stride 26, k0 even)
            v2f a; a.x = p[0]; a.y = p[1];
            afrag[kb] = a;
        }
        const int tokbase = blockIdx.x * TOK_PER_WG + tt * 16;
        #pragma unroll
        for (int nt = 0; nt < 4; ++nt) {
            v8f acc = {0.f, 0.f, 0.f, 0.f, 0.f, 0.f, 0.f, 0.f};
            #pragma unroll
            for (int kb = 0; kb < 6; ++kb) {
                int k0 = 4 * kb + 2 * lhalf;
                const float* q = &sB[nt * 16 + lmod][k0];
                v2f b; b.x = q[0]; b.y = q[1];
                acc = __builtin_amdgcn_wmma_f32_16x16x4_f32(
                          /*neg_a=*/false, afrag[kb],
                          /*neg_b=*/false, b,
                          /*c_mod=*/(short)0, acc,
                          /*reuse_a=*/false, /*reuse_b=*/false);
            }
            // C/D layout: VGPR r -> (M = r + 8*lhalf, N = lmod)
            const int chan = nt * 16 + lmod;
            float* obase = out + (size_t)(tokbase + 8 * lhalf) * Cc + chan;
            if (fullblock) {
                // Unguarded, non-temporal stores: output is write-once, never
                // re-read -> don't occupy L2 lines (store TH=NT).
                #pragma unroll
                for (int r = 0; r < 8; ++r)
                    __builtin_nontemporal_store(acc[r], obase + r * Cc);
            } else {
                #pragma unroll
                for (int r = 0; r < 8; ++r) {
                    int t = tokbase + r + 8 * lhalf;
                    if (t < BT) out[(size_t)t * Cc + chan] = acc[r];
                }
            }
        }
    }
}

extern "C" void kernel_launch(void* const* d_in, const int* in_sizes, int n_in,
                              void* d_out, int out_size, void* d_ws, size_t ws_size,
                              hipStream_t stream) {
    const float* x       = (const float*)d_in[0];   // (B,T)
    const float* ws      = (const float*)d_in[1];   // (S,C)
    const float* bs      = (const float*)d_in[2];   // (S,C)
    const float* loc_emb = (const float*)d_in[3];   // (L,C)
    const float* scales  = (const float*)d_in[4];   // (S,)
    const float* locs    = (const float*)d_in[5];   // (L,)
    float* out = (float*)d_out;                     // (B,T,C)

    const int BT = in_sizes[0];
    const int grid = (BT + TOK_PER_WG - 1) / TOK_PER_WG;
    mbe_wmma_kernel<<<grid, TOK_PER_WG, 0, stream>>>(x, ws, bs, loc_emb, scales, locs, out, BT);
}